// MeshPool_42966852829362
// MI455X (gfx1250) — compile-verified
//
#include <hip/hip_runtime.h>
#include <cstdint>

// MeshPool on MI455X (gfx1250, wave32).
// B=16 meshes, C=256 channels, E=9216 edges, K=4096 kept edges.
// Memory-bound (~290 MB total traffic -> ~12.5us @ 23.3 TB/s).
// Pass 1: score via all-ones-A f32 WMMA reduction (co-executes with loads).
// Pass 2: exact per-mesh radix-select top-K + stable compaction (LDS).
// Pass 3: gather with async-to-LDS index staging + NT streaming stores.

#define NB 16
#define NC 256
#define NE 9216
#define NK 4096

typedef float v2f __attribute__((ext_vector_type(2)));
typedef float v8f __attribute__((ext_vector_type(8)));

__device__ __forceinline__ unsigned f2o(float f) {
  // monotone float -> uint mapping (all valid scores >= 0 -> keys >= 0x80000000)
  unsigned u = __float_as_uint(f);
  return (u & 0x80000000u) ? ~u : (u | 0x80000000u);
}

__device__ __forceinline__ unsigned umin_u(unsigned a, unsigned b) { return a < b ? a : b; }

// ---------------------------------------------------------------------------
// Kernel 1: score[b,e] = sum_c x[b,c,e]^2, masked by edges_count, stored as
// orderable u32. One wave handles 16 edges; 64 chained V_WMMA_F32_16X16X4_F32
// with A == ones(16x4) give D[m][n] = sum over the 4 staged channels at edge n.
// Lane L supplies B values for column N = L&15; lane-half picks the channel
// pair, so the 4 K-slots hold channels {c0..c0+3} (sum is layout-invariant
// because A is all ones).
// ---------------------------------------------------------------------------
__global__ void __launch_bounds__(256) meshpool_score_kernel(
    const float* __restrict__ x, const int* __restrict__ edges_count,
    unsigned* __restrict__ score_u) {
  const int tid  = threadIdx.x;
  const int wave = tid >> 5;          // 8 waves / block
  const int lane = tid & 31;
  const int half = (lane >> 4) & 1;   // lane half selects channel pair
  const int b    = blockIdx.x / (NE / 128);
  const int tile = blockIdx.x % (NE / 128);
  const int e0   = tile * 128 + wave * 16;
  const int ec   = e0 + (lane & 15);

  const float* p = x + ((size_t)b * NC + 2 * half) * NE + ec;

  v2f ones; ones.x = 1.0f; ones.y = 1.0f;
  v8f acc = {0.f, 0.f, 0.f, 0.f, 0.f, 0.f, 0.f, 0.f};

#pragma unroll 4
  for (int c0 = 0; c0 < NC; c0 += 4) {
    float a0 = p[(size_t)c0 * NE];
    float a1 = p[(size_t)(c0 + 1) * NE];
    v2f bb; bb.x = a0 * a0; bb.y = a1 * a1;
    acc = __builtin_amdgcn_wmma_f32_16x16x4_f32(
        false, ones, false, bb, (short)0, acc, false, false);
  }

  // Every D row equals the column sum; lane L holds score for edge e0+(L&15).
  float s = acc[0];
  if (lane < 16) {
    const int e   = e0 + lane;
    const int cnt = edges_count[b];
    score_u[(size_t)b * NE + e] = (e < cnt) ? f2o(s) : 0u;
  }
}

// ---------------------------------------------------------------------------
// Kernel 2: per-mesh exact top-K threshold via 4-pass MSD radix select on the
// orderable keys (in LDS), then stable ascending-index compaction with
// lowest-index-first tie handling (matches jax.lax.top_k + sort).
// One block per mesh, 256 threads, 36 keys/thread.
// ---------------------------------------------------------------------------
__global__ void __launch_bounds__(256) meshpool_select_kernel(
    const unsigned* __restrict__ score_u, int* __restrict__ keep) {
  __shared__ unsigned keys[NE];
  __shared__ unsigned hist[256];
  __shared__ unsigned s_gt[256];
  __shared__ unsigned s_eq[256];
  __shared__ unsigned s_pref, s_want;

  const int tid = threadIdx.x;
  const int b   = blockIdx.x;

  for (int e = tid; e < NE; e += 256) keys[e] = score_u[(size_t)b * NE + e];
  if (tid == 0) { s_pref = 0u; s_want = NK; }
  __syncthreads();

  // MSD radix select: find the K-th largest key value T and the number of
  // ties at T that must be kept (s_want).
  for (int pass = 0; pass < 4; ++pass) {
    const int shift = 24 - 8 * pass;
    hist[tid] = 0u;
    __syncthreads();
    const unsigned pref   = s_pref;
    const unsigned himask = (pass == 0) ? 0u : (0xFFFFFFFFu << (shift + 8));
    for (int e = tid; e < NE; e += 256) {
      unsigned k = keys[e];
      if ((k & himask) == (pref & himask))
        atomicAdd(&hist[(k >> shift) & 255u], 1u);
    }
    __syncthreads();
    if (tid == 0) {
      unsigned cum = 0u, want = s_want;
      for (int d = 255; d >= 0; --d) {
        unsigned c = hist[d];
        if (cum + c >= want) {
          s_pref = pref | ((unsigned)d << shift);
          s_want = want - cum;
          break;
        }
        cum += c;
      }
    }
    __syncthreads();
  }

  const unsigned T       = s_pref;
  const unsigned tieKeep = s_want;  // first tieKeep ties (by index) are kept

  // Stable compaction: per-thread contiguous chunk counts, serial block scan,
  // then ordered emission. Output positions are ascending in edge index.
  const int e0c = tid * (NE / 256);  // 36 keys per thread
  unsigned gtc = 0u, eqc = 0u;
  for (int i = 0; i < NE / 256; ++i) {
    unsigned k = keys[e0c + i];
    gtc += (k > T);
    eqc += (k == T);
  }
  s_gt[tid] = gtc; s_eq[tid] = eqc;
  __syncthreads();
  if (tid == 0) {
    unsigned ag = 0u, ae = 0u;
    for (int t = 0; t < 256; ++t) {
      unsigned g = s_gt[t], q = s_eq[t];
      s_gt[t] = ag; s_eq[t] = ae;
      ag += g; ae += q;
    }
  }
  __syncthreads();
  unsigned gtp = s_gt[tid], eqp = s_eq[tid];
  int* kb = keep + (size_t)b * NK;
  for (int i = 0; i < NE / 256; ++i) {
    const int e = e0c + i;
    const unsigned k = keys[e];
    if (k > T) {
      kb[gtp + umin_u(eqp, tieKeep)] = e;
      gtp++;
    } else if (k == T) {
      if (eqp < tieKeep) kb[gtp + eqp] = e;
      eqp++;
    }
  }
}

// ---------------------------------------------------------------------------
// Kernel 3: out[b,c,k] = x[b,c,idx[b,k]]. Block = (mesh, 64-channel tile,
// 1024-k tile). Index tile (4 KB) staged via CDNA5 async copy to LDS
// (global_load_async_to_lds_b128, ASYNCcnt), cached in registers, reused
// across all 64 channels. Coalesced NT streaming stores for out.
// ---------------------------------------------------------------------------
__global__ void __launch_bounds__(256) meshpool_gather_kernel(
    const float* __restrict__ x, const int* __restrict__ keep,
    float* __restrict__ out) {
  __shared__ int s_idx[1024];
  const int tid = threadIdx.x;
  const int blk = blockIdx.x;
  const int b   = blk >> 4;
  const int rem = blk & 15;
  const int ct  = rem >> 2;  // channel tile (64 channels)
  const int kt  = rem & 3;   // k tile (1024 kept edges)

  // Async DMA of this block's 1024 indices into LDS: 16B per lane.
  const int* g = keep + (size_t)b * NK + kt * 1024 + 4 * tid;
  unsigned lds_off = (unsigned)(uintptr_t)(&s_idx[0]) + 16u * (unsigned)tid;
  asm volatile("global_load_async_to_lds_b128 %0, %1, off"
               :: "v"(lds_off), "v"((unsigned long long)(uintptr_t)g)
               : "memory");
  asm volatile("s_wait_asynccnt 0" ::: "memory");
  __syncthreads();

  const int i0 = s_idx[tid];
  const int i1 = s_idx[tid + 256];
  const int i2 = s_idx[tid + 512];
  const int i3 = s_idx[tid + 768];

  const float* xb = x   + ((size_t)b * NC + ct * 64) * NE;
  float*       ob = out + ((size_t)b * NC + ct * 64) * NK + kt * 1024;

  for (int cc = 0; cc < 64; ++cc) {
    const float* row  = xb + (size_t)cc * NE;
    float*       orow = ob + (size_t)cc * NK;
    __builtin_nontemporal_store(row[i0], &orow[tid]);
    __builtin_nontemporal_store(row[i1], &orow[tid + 256]);
    __builtin_nontemporal_store(row[i2], &orow[tid + 512]);
    __builtin_nontemporal_store(row[i3], &orow[tid + 768]);
  }
}

extern "C" void kernel_launch(void* const* d_in, const int* in_sizes, int n_in,
                              void* d_out, int out_size, void* d_ws, size_t ws_size,
                              hipStream_t stream) {
  (void)in_sizes; (void)n_in; (void)out_size; (void)ws_size;
  const float* x           = (const float*)d_in[0];
  const int*   edges_count = (const int*)d_in[1];  // JAX x64 off -> int32
  float*       out         = (float*)d_out;

  unsigned* score_u = (unsigned*)d_ws;                                   // B*E u32 = 576 KB
  int*      keep    = (int*)((char*)d_ws + (size_t)NB * NE * sizeof(unsigned)); // B*K i32 = 256 KB

  meshpool_score_kernel<<<dim3(NB * (NE / 128)), dim3(256), 0, stream>>>(
      x, edges_count, score_u);
  meshpool_select_kernel<<<dim3(NB), dim3(256), 0, stream>>>(score_u, keep);
  meshpool_gather_kernel<<<dim3(NB * 16), dim3(256), 0, stream>>>(x, keep, out);
}